// MultiHeadBinaryPositionIO_62380105007364
// MI455X (gfx1250) — compile-verified
//
#include <hip/hip_runtime.h>
#include <hip/hip_bf16.h>
#include <cstdint>
#include <cstddef>

typedef float v2f __attribute__((ext_vector_type(2)));
typedef float v8f __attribute__((ext_vector_type(8)));

#define NS 4096   // sequence
#define ND 1024   // model dim
#define NH 16     // heads
#define NBATCH 16 // batch
#define HD 64     // head dim
#define NBITS 12
#define KSPLIT 8  // K-chunks for the big GEMM (memory-level parallelism)

#define LOGEPS  (-18.420680743952367f)   // log(1e-8)
#define LOG1PE  (9.99999993922529e-09f)  // log1p(1e-8)

// d_out layout (floats): [0,128) char_value, [128,144) new_offset, [144,...) weights [B,H,1,S]
#define OUT_CHAR 0
#define OUT_NOFF 128
#define OUT_W    144

// workspace layout (floats)
#define WS_LS    0                              // B*S            = 65536
#define WS_MAX   65536                          // B              = 16
#define WS_TPART 65568                          // KSPLIT*B*H*D   = 8*262144
#define WS_TFIN  (WS_TPART + KSPLIT*262144)     // B*H*D          = 262144
#define WS_MULTI (WS_TFIN + 262144)             // B*D            = 16384
#define WS_OUT   (WS_MULTI + 16384)             // B*D            = 16384

// ---------------------------------------------------------------------------
// Kernel 1: per-(b,s) log-score via XOR+popcount, block max-reduce per batch.
// ---------------------------------------------------------------------------
__global__ void k_logscore(const int* __restrict__ positions,
                           const int* __restrict__ anchor,
                           const int* __restrict__ read_offset,
                           float* __restrict__ ls, float* __restrict__ maxls) {
  const int b = blockIdx.x;
  const int tid = threadIdx.x;
  const int anc = anchor[b];
  const int q = read_offset[b] & 0xFFF;
  float local = -INFINITY;
  for (int s = tid; s < NS; s += 256) {
    const int pos = positions[b * NS + s];
    float v = -INFINITY;
    if (pos > anc) {
      int rel = pos - anc - 1;
      rel = rel > 4095 ? 4095 : rel;            // rel >= 0 already (pos > anc)
      const int mism = __popc((rel ^ q) & 0xFFF);
      v = (float)(NBITS - mism) * LOG1PE + (float)mism * LOGEPS;
    }
    ls[b * NS + s] = v;
    local = fmaxf(local, v);
  }
  __shared__ float red[256];
  red[tid] = local;
  __syncthreads();
  for (int o = 128; o > 0; o >>= 1) {
    if (tid < o) red[tid] = fmaxf(red[tid], red[tid + o]);
    __syncthreads();
  }
  if (tid == 0) maxls[b] = red[0];
}

// ---------------------------------------------------------------------------
// Kernel 2: per-(b,h) softmax: Z reduction then normalized weights -> d_out.
// ---------------------------------------------------------------------------
__global__ void k_weights(const float* __restrict__ ls,
                          const float* __restrict__ maxls,
                          float* __restrict__ wout) {
  const int bh = blockIdx.x;
  const int b = bh >> 4, h = bh & 15;
  const int tid = threadIdx.x;
  const float scale = exp2f(-0.5f * (float)(h + 1)) * 10.0f;  // slope_h / T
  const float mx = maxls[b];
  const float* lsb = ls + (size_t)b * NS;
  float lsum = 0.0f;
  for (int s = tid; s < NS; s += 256) {
    const float v = lsb[s];
    lsum += (v == -INFINITY) ? 0.0f : expf(scale * (v - mx));
  }
  __shared__ float red[256];
  red[tid] = lsum;
  __syncthreads();
  for (int o = 128; o > 0; o >>= 1) {
    if (tid < o) red[tid] += red[tid + o];
    __syncthreads();
  }
  const float inv = 1.0f / red[0];
  float* w = wout + (size_t)bh * NS;
  for (int s = tid; s < NS; s += 256) {
    const float v = lsb[s];
    w[s] = (v == -INFINITY) ? 0.0f : expf(scale * (v - mx)) * inv;
  }
}

// ---------------------------------------------------------------------------
// Kernel 3 (the big one): t[b,h,d] = sum_s w[b,h,s] * x[b,s,d]
// GEMM per batch: A = weights [M=16 heads x K=4096], B = x [K=4096 x N=1024].
// grid = (ntile=16, b=16, kchunk=8); block = 128 (4 waves, 16 N-cols each).
// V_WMMA_F32_16X16X4_F32, streaming x once (memory-bound, 256 MB).
// Unroll 8 + 8-way K split => ~18 MB of loads in flight across the device,
// enough to cover HBM latency at 23.3 TB/s.
// ---------------------------------------------------------------------------
__global__ void k_gemm_tw(const float* __restrict__ x,
                          const float* __restrict__ w,
                          float* __restrict__ tpart) {
  const int b = blockIdx.y;
  const int kc = blockIdx.z;
  const int wv = threadIdx.x >> 5;
  const int lane = threadIdx.x & 31;
  const int half = lane >> 4;       // lane group 0/1
  const int l16 = lane & 15;
  const int n0 = blockIdx.x * 64 + wv * 16;

  // A row m = l16 (head); A[m,k] = w[b,m,k]
  const float* wp = w + ((size_t)b * NH + l16) * NS;
  // B col n = n0+l16; B[k,n] = x[b,k,n]
  const float* xp = x + (size_t)b * NS * ND + n0 + l16;

  v8f acc = {};
  const int kbeg = kc * (NS / KSPLIT);
  const int kend = kbeg + (NS / KSPLIT);
#pragma unroll 8
  for (int k = kbeg; k < kend; k += 4) {
    const int ka = k + 2 * half;                 // this lane-group's K rows
    v2f a = *(const v2f*)(wp + ka);              // b64: w[b,l16,ka], w[b,l16,ka+1]
    v2f bb;
    bb.x = xp[(size_t)ka * ND];                  // x[b,ka,  n]
    bb.y = xp[(size_t)(ka + 1) * ND];            // x[b,ka+1,n]
    acc = __builtin_amdgcn_wmma_f32_16x16x4_f32(false, a, false, bb,
                                                (short)0, acc, false, false);
  }
  // C layout: VGPR i -> row M = i + 8*half, col N = n0 + l16
  float* tp = tpart + (size_t)kc * (NBATCH * NH * ND);
#pragma unroll
  for (int i = 0; i < 8; ++i) {
    const int h = i + 8 * half;
    tp[((size_t)b * NH + h) * ND + n0 + l16] = acc[i];
  }
}

// ---------------------------------------------------------------------------
// Kernel 4: reduce the KSPLIT K-chunk partials of t.
// ---------------------------------------------------------------------------
__global__ void k_tred(const float* __restrict__ tpart, float* __restrict__ tfin) {
  const size_t i = (size_t)blockIdx.x * 256 + threadIdx.x;
  const size_t n = (size_t)NBATCH * NH * ND;
  float s = 0.0f;
#pragma unroll
  for (int c = 0; c < KSPLIT; ++c) s += tpart[i + (size_t)c * n];
  tfin[i] = s;
}

// ---------------------------------------------------------------------------
// Kernel 5: multi[b, h*64+e] = sum_d t[b,h,d] * Wv[h,e,d]
// Per head: A = t[:,h,:] [M=16 batch x K=1024], B = Wv[h]^T [K=1024 x N=64].
// grid = 16 heads; block = 128 (4 waves x 16 e-cols).
// ---------------------------------------------------------------------------
__global__ void k_gemm_hv(const float* __restrict__ tfin,
                          const float* __restrict__ Wv,
                          float* __restrict__ multi) {
  const int h = blockIdx.x;
  const int wv = threadIdx.x >> 5;
  const int lane = threadIdx.x & 31;
  const int half = lane >> 4;
  const int l16 = lane & 15;
  const int n0 = wv * 16;

  const float* ap = tfin + ((size_t)l16 * NH + h) * ND;        // A row m=b=l16
  const float* bp = Wv + ((size_t)h * HD + n0 + l16) * ND;     // B col n=e: Wv[h,e,:]

  v8f acc = {};
#pragma unroll 4
  for (int k = 0; k < ND; k += 4) {
    const int ka = k + 2 * half;
    v2f a = *(const v2f*)(ap + ka);
    v2f bb = *(const v2f*)(bp + ka);   // row-major along K -> b64
    acc = __builtin_amdgcn_wmma_f32_16x16x4_f32(false, a, false, bb,
                                                (short)0, acc, false, false);
  }
#pragma unroll
  for (int i = 0; i < 8; ++i) {
    const int bm = i + 8 * half;
    multi[(size_t)bm * ND + h * HD + n0 + l16] = acc[i];
  }
}

// ---------------------------------------------------------------------------
// Kernel 6: out[b,i] = sum_j multi[b,j] * Wo[i,j]
// A = multi [16 x 1024], B = Wo^T [1024 x 1024]. grid = 16 N-tiles of 64.
// ---------------------------------------------------------------------------
__global__ void k_gemm_o(const float* __restrict__ multi,
                         const float* __restrict__ Wo,
                         float* __restrict__ outb) {
  const int wv = threadIdx.x >> 5;
  const int lane = threadIdx.x & 31;
  const int half = lane >> 4;
  const int l16 = lane & 15;
  const int n0 = blockIdx.x * 64 + wv * 16;

  const float* ap = multi + (size_t)l16 * ND;          // A row m=b=l16
  const float* bp = Wo + (size_t)(n0 + l16) * ND;      // B col n=i: Wo[i,:]

  v8f acc = {};
#pragma unroll 4
  for (int k = 0; k < ND; k += 4) {
    const int ka = k + 2 * half;
    v2f a = *(const v2f*)(ap + ka);
    v2f bb = *(const v2f*)(bp + ka);
    acc = __builtin_amdgcn_wmma_f32_16x16x4_f32(false, a, false, bb,
                                                (short)0, acc, false, false);
  }
#pragma unroll
  for (int i = 0; i < 8; ++i) {
    const int bm = i + 8 * half;
    outb[(size_t)bm * ND + n0 + l16] = acc[i];
  }
}

// ---------------------------------------------------------------------------
// Kernel 7: char_value[b,c] = sum_j out[b,j] * Wc[c,j]; new_offset = ro + 1.
// ---------------------------------------------------------------------------
__global__ void k_final(const float* __restrict__ outb,
                        const float* __restrict__ Wc,
                        const int* __restrict__ read_offset,
                        float* __restrict__ dout) {
  const int tid = threadIdx.x;       // 128 threads: b = tid>>3, c = tid&7
  const int b = tid >> 3, c = tid & 7;
  const float* ob = outb + (size_t)b * ND;
  const float* wc = Wc + (size_t)c * ND;
  float s = 0.0f;
  for (int j = 0; j < ND; ++j) s += ob[j] * wc[j];
  dout[OUT_CHAR + tid] = s;
  if (tid < NBATCH) dout[OUT_NOFF + tid] = (float)(read_offset[tid] + 1);
}

// ---------------------------------------------------------------------------
extern "C" void kernel_launch(void* const* d_in, const int* in_sizes, int n_in,
                              void* d_out, int out_size, void* d_ws, size_t ws_size,
                              hipStream_t stream) {
  const float* x  = (const float*)d_in[0];
  const float* Wv = (const float*)d_in[1];
  const float* Wo = (const float*)d_in[2];
  const float* Wc = (const float*)d_in[3];
  const int* positions   = (const int*)d_in[4];
  const int* anchor      = (const int*)d_in[5];
  const int* read_offset = (const int*)d_in[6];
  float* out = (float*)d_out;
  float* ws  = (float*)d_ws;

  // 1) log-scores + per-batch max
  k_logscore<<<NBATCH, 256, 0, stream>>>(positions, anchor, read_offset,
                                         ws + WS_LS, ws + WS_MAX);
  // 2) softmax weights -> d_out weights slot (also A-matrix of the big GEMM)
  k_weights<<<NBATCH * NH, 256, 0, stream>>>(ws + WS_LS, ws + WS_MAX, out + OUT_W);
  // 3) t = weights @ x  (streams x once; K split 8 ways for MLP)
  k_gemm_tw<<<dim3(16, NBATCH, KSPLIT), 128, 0, stream>>>(x, out + OUT_W, ws + WS_TPART);
  // 4) fold K-chunk partials
  k_tred<<<(NBATCH * NH * ND) / 256, 256, 0, stream>>>(ws + WS_TPART, ws + WS_TFIN);
  // 5) multi = per-head t @ Wv^T
  k_gemm_hv<<<NH, 128, 0, stream>>>(ws + WS_TFIN, Wv, ws + WS_MULTI);
  // 6) out = multi @ Wo^T
  k_gemm_o<<<16, 128, 0, stream>>>(ws + WS_MULTI, Wo, ws + WS_OUT);
  // 7) char_value + new_offset
  k_final<<<1, 128, 0, stream>>>(ws + WS_OUT, Wc, read_offset, out);
}